// CEWald_3573412790705
// MI455X (gfx1250) — compile-verified
//
#include <hip/hip_runtime.h>
#include <hip/hip_bf16.h>
#include <math.h>
#include <stdint.h>

typedef __attribute__((ext_vector_type(2))) float v2f;
typedef __attribute__((ext_vector_type(4))) float f32x4;
typedef __attribute__((ext_vector_type(8))) float v8f;
typedef __attribute__((ext_vector_type(4))) unsigned int u32x4;
typedef __attribute__((ext_vector_type(4))) int i32x4;
typedef __attribute__((ext_vector_type(8))) int i32x8;

// LDS-qualified volatile views: volatile prevents the optimizer from folding
// reads of TDM-written memory to undef; address_space(3) keeps the accesses on
// the DS pipe (ds_load_b32/b128) instead of volatile flat loads at SYS scope.
typedef volatile __attribute__((address_space(3))) float lds_vf;
typedef volatile __attribute__((address_space(3))) f32x4 lds_vf4;

#define TWO_PI_F        6.28318530717958647692f
#define ONE_OVER_SQRTPI 0.56418958354775628695f
#define ALPHA_F         (4.0f / 10.0f + 0.001f)
#define ALPHA2_F        (ALPHA_F * ALPHA_F)
#define ON_CUT_F        2.5f
#define OFF_CUT_F       7.5f
#define EPS_F           1e-8f
#define APB             512   // atoms per batch in this problem (16384/32)

// ---------------------------------------------------------------------------
// TDM: 1-D contiguous f32 copy global -> LDS via TENSOR_LOAD_TO_LDS.
// D# built per CDNA5 ISA ch.8: group0 = {count=1 | lds_addr | global_addr |
// type=2}, group1 = {data_size=4B, tensor_dim0=n, tensor_dim1=1, tile_dim0=n}.
// Issued once per wave (EXEC ignored for tensor ops); completion tracked with
// TENSORcnt.  Arity of the builtin differs per toolchain; the TDM header only
// ships with the 6-arg toolchain, so dispatch on its presence.
// ---------------------------------------------------------------------------
__device__ __forceinline__ void tdm_copy_1d_f32(const void* gsrc,
                                                unsigned lds_byte_off,
                                                int nElem) {
    unsigned long long ga = (unsigned long long)(size_t)gsrc;
    u32x4 g0;
    g0.x = 1u;                                    // count=1, user descriptor
    g0.y = lds_byte_off;                          // lds_addr (bytes)
    g0.z = (unsigned)(ga & 0xFFFFFFFFull);        // global_addr[31:0]
    g0.w = (unsigned)((ga >> 32) & 0x1FFFFFFull)  // global_addr[56:32]
         | (2u << 30);                            // type = 2 ("image")
    i32x8 g1;
    g1[0] = (2 << 16);                                        // data_size = 4B
    g1[1] = (int)(((unsigned)nElem & 0xFFFFu) << 16);         // tensor_dim0[15:0]
    g1[2] = (int)(((unsigned)nElem >> 16) & 0xFFFFu)          // tensor_dim0[31:16]
          | (1 << 16);                                        // tensor_dim1 = 1
    g1[3] = (int)(((unsigned)nElem & 0xFFFFu) << 16);         // tile_dim0 = n
    g1[4] = 0;                                                // tile_dim1/2 unused
    g1[5] = nElem;                                            // dim0_stride[31:0]
    g1[6] = 0;                                                // stride hi bits
    g1[7] = 0;
    i32x4 z4 = {0, 0, 0, 0};
#if __has_include(<hip/amd_detail/amd_gfx1250_TDM.h>)
    i32x8 z8 = {0, 0, 0, 0, 0, 0, 0, 0};
    __builtin_amdgcn_tensor_load_to_lds(g0, g1, z4, z4, z8, 0);
#else
    __builtin_amdgcn_tensor_load_to_lds(g0, g1, z4, z4, 0);
#endif
}

// ---------------------------------------------------------------------------
// Kernel 0: zero the output accumulator (harness poisons d_out with 0xAA).
// ---------------------------------------------------------------------------
__global__ void zero_f32(float* p, int n) {
    int i = blockIdx.x * blockDim.x + threadIdx.x;
    if (i < n) p[i] = 0.0f;
}

// ---------------------------------------------------------------------------
// Kernel 1: per (batch, k) precompute k-vectors (padded to 4 floats for the
// 16x16x4 WMMA K-dim) and qg = exp(-k^2/(4a^2))/k^2.  Padded slots zeroed.
// ---------------------------------------------------------------------------
__global__ void precompute_k(const float* __restrict__ cell,
                             const float* __restrict__ kmul,
                             float* __restrict__ kvec,  // [B][Kpad][4]
                             float* __restrict__ qg,    // [B][Kpad]
                             int B, int K, int Kpad) {
    int t = blockIdx.x * blockDim.x + threadIdx.x;
    if (t >= B * Kpad) return;
    int b = t / Kpad;
    int kk = t - b * Kpad;
    float kx = 0.0f, ky = 0.0f, kz = 0.0f, g = 0.0f;
    if (kk < K) {
        float Lx = cell[b * 9 + 0];
        float Ly = cell[b * 9 + 4];
        float Lz = cell[b * 9 + 8];
        kx = TWO_PI_F * kmul[kk * 3 + 0] / Lx;
        ky = TWO_PI_F * kmul[kk * 3 + 1] / Ly;
        kz = TWO_PI_F * kmul[kk * 3 + 2] / Lz;
        float k2 = kx * kx + ky * ky + kz * kz;
        g = __expf(-0.25f * k2 / ALPHA2_F) / k2;
    }
    kvec[t * 4 + 0] = kx;
    kvec[t * 4 + 1] = ky;
    kvec[t * 4 + 2] = kz;
    kvec[t * 4 + 3] = 0.0f;
    qg[t] = g;
}

// ---------------------------------------------------------------------------
// Kernel 2: real-space pair sum with wave32 segmented scan on sorted idx_i;
// only segment tails issue global_atomic_add_f32 (~32x fewer atomics).
// ---------------------------------------------------------------------------
__global__ void real_space(const float* __restrict__ Qa,
                           const float* __restrict__ rij,
                           const int* __restrict__ idx_i,
                           const int* __restrict__ idx_j,
                           float* __restrict__ out,
                           int nPairs) {
    int p = blockIdx.x * blockDim.x + threadIdx.x;
    int key = -1;
    float v = 0.0f;
    if (p < nPairs) {
        int i = idx_i[p];
        int j = idx_j[p];
        float r = rij[p];
        float fac = Qa[i] * Qa[j];
        float x = (r - ON_CUT_F) * (1.0f / (OFF_CUT_F - ON_CUT_F));
        float f;
        if (x <= 0.0f) {
            f = 1.0f;
        } else if (x >= 1.0f) {
            f = 0.0f;
        } else {
            float fp = __expf(-1.0f / x);
            float fm = __expf(-1.0f / (1.0f - x));
            f = fm / (fp + fm);
        }
        float coulomb = 1.0f / r;
        float damped  = rsqrtf(r * r + 1.0f);
        float pw = fac * (f * damped + (1.0f - f) * coulomb) * erfcf(ALPHA_F * r);
        key = i;
        v = pw;
    }
    int lane = (int)(threadIdx.x & 31u);
    #pragma unroll
    for (int off = 1; off < 32; off <<= 1) {
        float ov = __shfl_up(v, off, 32);
        int   ok = __shfl_up(key, off, 32);
        if (lane >= off && ok == key) v += ov;
    }
    int nk = __shfl_down(key, 1, 32);
    if (key >= 0 && (lane == 31 || nk != key)) {
        atomicAdd(&out[key], v);
    }
}

// ---------------------------------------------------------------------------
// Shared WMMA fragment helper for the structure-factor kernels.
// A (16x4):  lanes 0-15 hold {x,y} of atom M=lane%16; lanes 16-31 hold {z,0}
// B (4x16):  lanes 0-15 hold rows {0,1} of kvec col N=lane%16; hi half rows {2,3}
// D (16x16): VGPR r -> M = r + 8*hi, N = lane%16
// ---------------------------------------------------------------------------
__device__ __forceinline__ v8f phase_wmma(v2f afrag, v2f bfrag) {
    v8f c = {};
    return __builtin_amdgcn_wmma_f32_16x16x4_f32(
        /*neg_a=*/false, afrag, /*neg_b=*/false, bfrag,
        /*c_mod=*/(short)0, c, /*reuse_a=*/false, /*reuse_b=*/false);
}

// ---------------------------------------------------------------------------
// Kernel 3 (TDM path): one block = (batch b, 8 k-tiles).  Wave 0 stages the
// batch's R (APB*3 f32) and Qa (APB f32) into LDS with TENSOR_LOAD_TO_LDS,
// waits TENSORcnt, barrier; then each wave owns one 16-kvec tile and loops
// the 32 atom-tiles: WMMA phase matrix -> sincos -> charge-weighted
// accumulation, shfl_xor(16) fold, store.
//
// NOTE: the TDM builtin carries no pointer to smem, so the optimizer sees no
// store to the array and would fold plain LDS reads to undef (observed in
// round-2 disasm).  All LDS reads therefore go through volatile AS(3)
// pointers, forcing real ds_load_b32/b128 on the DS pipe.
// ---------------------------------------------------------------------------
__global__ void structure_factor_tdm(const float* __restrict__ Qa,
                                     const float* __restrict__ R,
                                     const float* __restrict__ kvec,
                                     float* __restrict__ q_real,
                                     float* __restrict__ q_imag,
                                     int Kpad, int nkt, int ktBlocks) {
    __shared__ float smem[APB * 4];  // [0,APB*3): R raw ; [APB*3,APB*4): Qa

    int b   = blockIdx.x / ktBlocks;
    int kt0 = (blockIdx.x - b * ktBlocks) * 8;

    if (threadIdx.x < 32) {  // wave 0 issues the two TDM descriptors
        tdm_copy_1d_f32(R + (size_t)b * APB * 3, 0u, APB * 3);
        tdm_copy_1d_f32(Qa + (size_t)b * APB, (unsigned)(APB * 3 * sizeof(float)), APB);
        __builtin_amdgcn_s_wait_tensorcnt(0);
    }
    __syncthreads();

    int wave = (int)(threadIdx.x >> 5);
    int kt = kt0 + wave;
    if (kt >= nkt) return;  // uniform per wave, after the barrier

    int lane = (int)(threadIdx.x & 31u);
    int n    = lane & 15;
    int hi   = lane >> 4;

    const float* kv = &kvec[((size_t)b * Kpad + kt * 16 + n) * 4];
    v2f bfrag;
    bfrag.x = kv[hi ? 2 : 0];
    bfrag.y = kv[hi ? 3 : 1];

    // volatile AS(3) views of the TDM-staged LDS data (see NOTE above)
    lds_vf*  Rs  = (lds_vf*)smem;
    lds_vf4* Qs4 = (lds_vf4*)(smem + APB * 3);

    float accR = 0.0f, accI = 0.0f;
    for (int t = 0; t < APB / 16; ++t) {
        int m = t * 16 + n;
        v2f afrag;
        if (hi) { afrag.x = Rs[m * 3 + 2]; afrag.y = 0.0f; }
        else    { afrag.x = Rs[m * 3 + 0]; afrag.y = Rs[m * 3 + 1]; }

        v8f d = phase_wmma(afrag, bfrag);

        // 8 contiguous charges for this lane's atom group: two b128 LDS loads
        f32x4 qa = Qs4[t * 4 + hi * 2 + 0];
        f32x4 qb = Qs4[t * 4 + hi * 2 + 1];
        float qv[8] = {qa.x, qa.y, qa.z, qa.w, qb.x, qb.y, qb.z, qb.w};

        #pragma unroll
        for (int r = 0; r < 8; ++r) {
            float s, co;
            __sincosf(d[r], &s, &co);
            accR += qv[r] * co;
            accI += qv[r] * s;
        }
    }
    accR += __shfl_xor(accR, 16, 32);
    accI += __shfl_xor(accI, 16, 32);
    if (!hi) {
        size_t o = (size_t)b * Kpad + kt * 16 + n;
        q_real[o] = accR;
        q_imag[o] = accI;
    }
}

// ---------------------------------------------------------------------------
// Kernel 3 (fallback, global loads): one wave per (batch, k-tile).
// ---------------------------------------------------------------------------
__global__ void structure_factor(const float* __restrict__ Qa,
                                 const float* __restrict__ R,
                                 const float* __restrict__ kvec,
                                 float* __restrict__ q_real,
                                 float* __restrict__ q_imag,
                                 int atomsPerBatch, int Kpad, int nkt,
                                 int totalWaves) {
    int waveId = blockIdx.x * (blockDim.x >> 5) + (threadIdx.x >> 5);
    if (waveId >= totalWaves) return;
    int b  = waveId / nkt;
    int kt = waveId - b * nkt;

    int lane = (int)(threadIdx.x & 31u);
    int n    = lane & 15;
    int hi   = lane >> 4;

    const float* kv = &kvec[((size_t)b * Kpad + kt * 16 + n) * 4];
    v2f bfrag;
    bfrag.x = kv[hi ? 2 : 0];
    bfrag.y = kv[hi ? 3 : 1];

    float accR = 0.0f, accI = 0.0f;
    int tiles = atomsPerBatch >> 4;
    int base  = b * atomsPerBatch;

    for (int t = 0; t < tiles; ++t) {
        int m0 = base + t * 16;
        const float* Rp = &R[(size_t)(m0 + n) * 3];
        v2f afrag;
        if (hi) { afrag.x = Rp[2]; afrag.y = 0.0f; }
        else    { afrag.x = Rp[0]; afrag.y = Rp[1]; }

        v8f d = phase_wmma(afrag, bfrag);

        #pragma unroll
        for (int r = 0; r < 8; ++r) {
            float q = Qa[m0 + r + (hi ? 8 : 0)];
            float s, co;
            __sincosf(d[r], &s, &co);
            accR += q * co;
            accI += q * s;
        }
    }
    accR += __shfl_xor(accR, 16, 32);
    accI += __shfl_xor(accI, 16, 32);
    if (!hi) {
        size_t o = (size_t)b * Kpad + kt * 16 + n;
        q_real[o] = accR;
        q_imag[o] = accI;
    }
}

// ---------------------------------------------------------------------------
// Kernel 4: per-batch reductions: e_recip_b = 2pi/V * sum_k qg*(qr^2+qi^2),
// wnorm = sum_atoms (q^2 + eps).  One block per batch.
// ---------------------------------------------------------------------------
__global__ void batch_reduce(const float* __restrict__ Qa,
                             const float* __restrict__ cell,
                             const float* __restrict__ qg,
                             const float* __restrict__ q_real,
                             const float* __restrict__ q_imag,
                             float* __restrict__ e_recip_b,
                             float* __restrict__ wnorm,
                             int K, int Kpad, int atomsPerBatch) {
    __shared__ float sm[256];
    int b = blockIdx.x;
    int tid = threadIdx.x;

    float s = 0.0f;
    for (int kk = tid; kk < K; kk += 256) {
        size_t o = (size_t)b * Kpad + kk;
        float qr = q_real[o], qi = q_imag[o];
        s += qg[o] * (qr * qr + qi * qi);
    }
    sm[tid] = s;
    __syncthreads();
    for (int st = 128; st > 0; st >>= 1) {
        if (tid < st) sm[tid] += sm[tid + st];
        __syncthreads();
    }
    if (tid == 0) {
        float vol = cell[b * 9 + 0] * cell[b * 9 + 4] * cell[b * 9 + 8];
        e_recip_b[b] = (TWO_PI_F / vol) * sm[0];
    }
    __syncthreads();

    float w = 0.0f;
    for (int a = tid; a < atomsPerBatch; a += 256) {
        float q = Qa[b * atomsPerBatch + a];
        w += q * q + EPS_F;
    }
    sm[tid] = w;
    __syncthreads();
    for (int st = 128; st > 0; st >>= 1) {
        if (tid < st) sm[tid] += sm[tid + st];
        __syncthreads();
    }
    if (tid == 0) wnorm[b] = sm[0];
}

// ---------------------------------------------------------------------------
// Kernel 5: per-atom finalize: out += w * e_recip_b[b] - alpha/sqrt(pi)*q^2
// ---------------------------------------------------------------------------
__global__ void finalize(const float* __restrict__ Qa,
                         const int* __restrict__ batch_seg,
                         const float* __restrict__ e_recip_b,
                         const float* __restrict__ wnorm,
                         float* __restrict__ out,
                         int nAtoms) {
    int i = blockIdx.x * blockDim.x + threadIdx.x;
    if (i >= nAtoms) return;
    int b = batch_seg[i];
    float q = Qa[i];
    float q2 = q * q;
    float w = (q2 + EPS_F) / wnorm[b];
    out[i] += w * e_recip_b[b] - ALPHA_F * ONE_OVER_SQRTPI * q2;
}

// ---------------------------------------------------------------------------
extern "C" void kernel_launch(void* const* d_in, const int* in_sizes, int n_in,
                              void* d_out, int out_size, void* d_ws, size_t ws_size,
                              hipStream_t stream) {
    const float* Qa       = (const float*)d_in[0];
    const float* rij      = (const float*)d_in[1];
    const float* R        = (const float*)d_in[2];
    const float* cell     = (const float*)d_in[3];
    const float* kmul     = (const float*)d_in[4];
    const int*   idx_i    = (const int*)d_in[5];
    const int*   idx_j    = (const int*)d_in[6];
    const int*   batchseg = (const int*)d_in[7];

    int nAtoms = in_sizes[0];
    int nPairs = in_sizes[1];
    int B      = in_sizes[3] / 9;
    int K      = in_sizes[4] / 3;
    int Kpad   = (K + 15) & ~15;
    int apb    = nAtoms / B;   // atoms per batch (contiguous by construction)

    // workspace layout (floats)
    float* ws     = (float*)d_ws;
    float* kvec   = ws;                                 // B*Kpad*4
    float* qg     = kvec + (size_t)B * Kpad * 4;        // B*Kpad
    float* qr     = qg   + (size_t)B * Kpad;            // B*Kpad
    float* qi     = qr   + (size_t)B * Kpad;            // B*Kpad
    float* erec   = qi   + (size_t)B * Kpad;            // B
    float* wnorm  = erec + B;                           // B

    float* out = (float*)d_out;

    zero_f32<<<(nAtoms + 255) / 256, 256, 0, stream>>>(out, nAtoms);

    precompute_k<<<(B * Kpad + 255) / 256, 256, 0, stream>>>(
        cell, kmul, kvec, qg, B, K, Kpad);

    real_space<<<(nPairs + 255) / 256, 256, 0, stream>>>(
        Qa, rij, idx_i, idx_j, out, nPairs);

    int nkt = Kpad / 16;
    if (apb == APB) {
        int ktBlocks = (nkt + 7) / 8;
        structure_factor_tdm<<<B * ktBlocks, 256, 0, stream>>>(
            Qa, R, kvec, qr, qi, Kpad, nkt, ktBlocks);
    } else {
        int totalWaves = B * nkt;
        structure_factor<<<(totalWaves + 7) / 8, 256, 0, stream>>>(
            Qa, R, kvec, qr, qi, apb, Kpad, nkt, totalWaves);
    }

    batch_reduce<<<B, 256, 0, stream>>>(
        Qa, cell, qg, qr, qi, erec, wnorm, K, Kpad, apb);

    finalize<<<(nAtoms + 255) / 256, 256, 0, stream>>>(
        Qa, batchseg, erec, wnorm, out, nAtoms);
}